// LeakageNlinCore_79620103733361
// MI455X (gfx1250) — compile-verified
//
#include <hip/hip_runtime.h>

typedef __attribute__((ext_vector_type(2))) float v2f;
typedef __attribute__((ext_vector_type(8))) float v8f;

#define N_CH     64
#define KDIM     128            // input features: fr(64) ++ fi(64)
#define NDIM     128            // output features: (c,o) flattened
#define ROWS     128            // rows (samples) per block = 8 waves * 16
#define FPAD     132            // padded row stride for F tile in LDS:
                                // bank = (4r + k) % 64; the two K-half lane
                                // groups use disjoint (mod 4) bank classes
#define NKSTEP   (KDIM / 4)     // 32 K-steps of the 16x16x4 f32 WMMA
#define NTILES   (NDIM / 16)    // 8 N-tiles
#define BFRAG_FLOATS (NKSTEP * NTILES * 32 * 2)   // 16384 floats = 64 KB

// ---------------------------------------------------------------------------
// Expand W[C,2,126] (per-channel leave-one-out linear) into the dense
// 128x128 matrix with zeros on the own-channel slots, written DIRECTLY in
// the per-lane WMMA B-fragment layout:
//   Bfrag[kk][nt][lane] = { Wfull[k0][n], Wfull[k0+1][n] }  (float2)
//   k0 = 4*kk + 2*(lane>=16),  n = 16*nt + lane%16
// so the main kernel's B loads are single contiguous ds_load_b64 per lane,
// bank-conflict-free, with no VGPR repacking.
// Row k of Wfull: k<64 -> fr of channel k, k>=64 -> fi of channel k-64.
// Col n = 2*c + o.
// ---------------------------------------------------------------------------
__global__ void build_wfrag_kernel(const float* __restrict__ W,
                                   float* __restrict__ bfrag) {
    int idx  = blockIdx.x * blockDim.x + threadIdx.x;   // 0 .. 16383
    int comp = idx & 1;                                 // 0: b.x, 1: b.y
    int p    = idx >> 1;                                // pair id
    int lane = p & 31;
    int nt   = (p >> 5) & 7;
    int kk   = p >> 8;

    int koff = (lane >> 4) * 2;
    int k = kk * 4 + koff + comp;
    int n = nt * 16 + (lane & 15);

    int c = n >> 1;
    int o = n & 1;
    int j = (k < N_CH) ? k : k - N_CH;
    float v = 0.0f;
    if (j != c) {
        int p_loo = (j < c) ? j : j - 1;                // position in LOO list
        int feat  = (k < N_CH) ? p_loo : 63 + p_loo;    // [fr_loo ++ fi_loo]
        v = W[c * 252 + o * 126 + feat];                // W[c][o][feat]
    }
    bfrag[idx] = v;
}

// ---------------------------------------------------------------------------
// Main kernel: out[32768 x 128] = F[32768 x 128] * Wfull[128 x 128]
// where F is computed on the fly from x: amp = xr^2 + xi^2, F = amp * x.
// fp32 WMMA (16x16x4) keeps reference precision; kernel is HBM-bound
// (~33.5 MB traffic, AI ~ 32 FLOP/B, floor ~1.4 us at 23.3 TB/s).
// ---------------------------------------------------------------------------
__global__ __launch_bounds__(256) void leakage_wmma_kernel(
        const float* __restrict__ x,       // [rows, 64, 2]
        const float* __restrict__ bfrag,   // swizzled B fragments, 16384 f32
        float* __restrict__ out) {         // [rows, 128]
    extern __shared__ float lds[];
    float* Ftile = lds;                    // ROWS * FPAD floats
    float* Bw    = lds + ROWS * FPAD;      // BFRAG_FLOATS floats

    const int tid      = threadIdx.x;
    const int row_base = blockIdx.x * ROWS;

    // ---- Stage swizzled B fragments into LDS (pure linear float4 copy) ----
    {
        const float4* src = (const float4*)bfrag;
        float4*       dst = (float4*)Bw;
        #pragma unroll
        for (int i = 0; i < BFRAG_FLOATS / 4 / 256; ++i)   // 16 iters
            dst[tid + i * 256] = src[tid + i * 256];
    }

    // ---- Stage x tile, apply cubic nonlinearity, build F in LDS ----
    {
        const float2* x2 = (const float2*)x;   // one (xr, xi) pair per channel
        #pragma unroll
        for (int i = 0; i < 32; ++i) {
            int lin = tid + i * 256;           // 0..8191 = 128 rows * 64 ch
            int r   = lin >> 6;
            int c   = lin & 63;
            float2 v = x2[(size_t)(row_base + r) * N_CH + c];
            float amp = v.x * v.x + v.y * v.y;
            Ftile[r * FPAD + c]        = amp * v.x;   // fr
            Ftile[r * FPAD + N_CH + c] = amp * v.y;   // fi
        }
    }

    __syncthreads();

    // ---- WMMA: each wave computes a 16x128 output tile ----
    const int wave    = tid >> 5;
    const int lane    = tid & 31;
    const int halfsel = lane >> 4;         // 0: lanes 0-15, 1: lanes 16-31
    const int lrow    = lane & 15;
    const int koff    = halfsel * 2;       // A layout: K = {0,1} / {2,3} halves

    v8f acc[NTILES] = {};                  // 8 N-tiles of 16 columns each

    const float* Frow  = &Ftile[(wave * 16 + lrow) * FPAD];
    const float* Blane = &Bw[lane * 2];    // this lane's fragment stream

    #pragma unroll 4
    for (int kk = 0; kk < NKSTEP; ++kk) {
        // A fragment (16x4 f32): VGPR0 = K=k0, VGPR1 = K=k0+1, contiguous b64
        v2f a = *(const v2f*)&Frow[kk * 4 + koff];
        #pragma unroll
        for (int nt = 0; nt < NTILES; ++nt) {
            // B fragment: one contiguous, conflict-free ds_load_b64
            v2f b = *(const v2f*)&Blane[(kk * NTILES + nt) * 64];
            acc[nt] = __builtin_amdgcn_wmma_f32_16x16x4_f32(
                false, a, false, b, (short)0, acc[nt], false, false);
        }
    }

    // ---- Store C/D fragments: VGPR v -> M = v + 8*halfsel, N = lrow ----
    // Lanes 0-15 of each instruction write one contiguous 64B run; lanes
    // 16-31 write the matching run 8 rows down.
    #pragma unroll
    for (int nt = 0; nt < NTILES; ++nt) {
        #pragma unroll
        for (int v = 0; v < 8; ++v) {
            int m = v + halfsel * 8;
            size_t r = (size_t)(row_base + wave * 16 + m);
            out[r * NDIM + nt * 16 + lrow] = acc[nt][v];
        }
    }
}

extern "C" void kernel_launch(void* const* d_in, const int* in_sizes, int n_in,
                              void* d_out, int out_size, void* d_ws, size_t ws_size,
                              hipStream_t stream) {
    const float* x = (const float*)d_in[0];   // [B,S,64,2] fp32
    const float* W = (const float*)d_in[1];   // [64,2,126] fp32
    float* out   = (float*)d_out;             // [B,S,64,2] fp32
    float* bfrag = (float*)d_ws;              // 64 KB swizzled-weight scratch

    // 1) Expand LOO weights into dense, B-fragment-swizzled form (rebuilt
    //    every call; d_ws carries no state across calls).
    build_wfrag_kernel<<<BFRAG_FLOATS / 256, 256, 0, stream>>>(W, bfrag);

    // 2) GEMM with on-the-fly cubic nonlinearity.
    int rows = in_sizes[0] / (2 * N_CH);      // B*S = 32768
    int grid = rows / ROWS;                   // 256 blocks
    size_t shmem = (size_t)(ROWS * FPAD + BFRAG_FLOATS) * sizeof(float);
    leakage_wmma_kernel<<<grid, 256, shmem, stream>>>(x, bfrag, out);
}